// GCPNetModel_80229989089898
// MI455X (gfx1250) — compile-verified
//
#include <hip/hip_runtime.h>
#include <hip/hip_bf16.h>

typedef __attribute__((ext_vector_type(2))) float v2f;
typedef __attribute__((ext_vector_type(8))) float v8f;

#define WAVES 4
#define MSTR 164   // padded row stride for merged[16][160] (bank-conflict free)
#define VSTR 52    // padded row stride for vec/vh staging [16][48]
#define GSTR 17    // padded row stride for gate [16][16]
#define KPAD 160   // 153 padded to multiple of 4

// ---------------- kernel 0: build WMMA B operands (K-pair interleaved) -----
// Btp_scalar[p][j][2] : p = k/2 in 0..79, j in 0..127, last dim = k&1
// Btp_gate  [p][vo][2]: p = k/2 in 0..63, vo in 0..15
__global__ void gcp_prep_kernel(const float* __restrict__ W_scalar,
                                const float* __restrict__ W_gate,
                                float* __restrict__ Btp_scalar,
                                float* __restrict__ Btp_gate) {
  int i = blockIdx.x * blockDim.x + threadIdx.x;
  if (i < KPAD * 128) {
    int k = i >> 7, j = i & 127;
    float v = (k < 153) ? W_scalar[j * 153 + k] : 0.0f;
    Btp_scalar[(k >> 1) * 256 + j * 2 + (k & 1)] = v;
  }
  if (i < 128 * 16) {
    int k = i >> 4, vo = i & 15;
    Btp_gate[(k >> 1) * 32 + vo * 2 + (k & 1)] = W_gate[vo * 128 + k];
  }
}

// ---------------- kernel 1: vdf = v_pre @ W_frames^T, zero accumulators ----
__global__ void gcp_vdf_kernel(const float* __restrict__ vector_rep,
                               const float* __restrict__ W_frames,
                               float* __restrict__ vdf,   // [N][3][3] (t,f)
                               float* __restrict__ sums,  // [N][9]
                               float* __restrict__ cnt,   // [N]
                               int N) {
  int n = blockIdx.x * blockDim.x + threadIdx.x;
  if (n >= N) return;
  const float* v = vector_rep + (size_t)n * 48;  // [V=16][3]
  float acc[9];
  #pragma unroll
  for (int i = 0; i < 9; ++i) acc[i] = 0.0f;
  #pragma unroll
  for (int vv = 0; vv < 16; ++vv) {
    float x = v[vv * 3 + 0], y = v[vv * 3 + 1], z = v[vv * 3 + 2];
    #pragma unroll
    for (int f = 0; f < 3; ++f) {
      float w = W_frames[f * 16 + vv];
      acc[0 * 3 + f] += x * w;
      acc[1 * 3 + f] += y * w;
      acc[2 * 3 + f] += z * w;
    }
  }
  #pragma unroll
  for (int i = 0; i < 9; ++i) vdf[(size_t)n * 9 + i] = acc[i];
  #pragma unroll
  for (int i = 0; i < 9; ++i) sums[(size_t)n * 9 + i] = 0.0f;
  cnt[n] = 0.0f;
}

// ---------------- kernel 2: per-edge frame matmul + atomic scatter ---------
__global__ void gcp_edge_kernel(const float* __restrict__ frames,
                                const long long* __restrict__ row_idx,
                                const float* __restrict__ vdf,
                                float* __restrict__ sums,
                                float* __restrict__ cnt,
                                int E) {
  int e = blockIdx.x * blockDim.x + threadIdx.x;
  if (e >= E) return;
  long long r = row_idx[e];
  const float* fr = frames + (size_t)e * 9;   // [s][t]
  const float* vd = vdf + (size_t)r * 9;      // [t][f]
  float f0 = fr[0], f1 = fr[1], f2 = fr[2];
  float f3 = fr[3], f4 = fr[4], f5 = fr[5];
  float f6 = fr[6], f7 = fr[7], f8 = fr[8];
  float* sb = sums + (size_t)r * 9;
  #pragma unroll
  for (int f = 0; f < 3; ++f) {
    float a = vd[0 + f], b = vd[3 + f], c = vd[6 + f];
    // m[s][f] = sum_t frames[s][t]*vdf[t][f]; local index = f*3+s
    atomicAdd(&sb[f * 3 + 0], f0 * a + f1 * b + f2 * c);
    atomicAdd(&sb[f * 3 + 1], f3 * a + f4 * b + f5 * c);
    atomicAdd(&sb[f * 3 + 2], f6 * a + f7 * b + f8 * c);
  }
  atomicAdd(&cnt[r], 1.0f);
}

// ---------------- kernel 3: fused node pass with f32 WMMA ------------------
__global__ __launch_bounds__(32 * WAVES)
void gcp_node_kernel(const float* __restrict__ scalar_rep,
                     const float* __restrict__ vector_rep,
                     const float* __restrict__ W_down,
                     const float* __restrict__ b_scalar,
                     const float* __restrict__ W_up,
                     const float* __restrict__ b_gate,
                     const float* __restrict__ sums,
                     const float* __restrict__ cnt,
                     const float* __restrict__ Btp_scalar,
                     const float* __restrict__ Btp_gate,
                     float* __restrict__ s_final,
                     float* __restrict__ v_final,
                     int N) {
  __shared__ __align__(16) float wd_lds[256];
  __shared__ __align__(16) float wu_lds[256];
  __shared__ __align__(16) float merged_lds[WAVES][16 * MSTR];
  __shared__ __align__(16) float vstage[WAVES][16 * VSTR];
  __shared__ __align__(16) float gate_lds[WAVES][16 * GSTR];

  const int tid = threadIdx.x;
  const int wave = tid >> 5;
  const int lane = tid & 31;
  const int lmod = lane & 15;
  const int lhalf = lane >> 4;

  for (int i = tid; i < 256; i += 32 * WAVES) {
    wd_lds[i] = W_down[i];
    wu_lds[i] = W_up[i];
  }

  const int base = (blockIdx.x * WAVES + wave) * 16;
  const bool active = (base + 16 <= N);   // wave-uniform (N % 16 == 0)
  float* M = merged_lds[wave];
  float* V = vstage[wave];
  float* G = gate_lds[wave];

  // -------- Phase A: build merged[16][160] in LDS --------
  if (active) {
    const float4* sr = (const float4*)(scalar_rep + (size_t)base * 128);
    #pragma unroll
    for (int it = 0; it < 16; ++it) {
      int idx = lane + 32 * it;           // 0..511
      int n = idx >> 5, c4 = idx & 31;
      float4 v = sr[n * 32 + c4];
      *(float4*)(M + n * MSTR + c4 * 4) = v;
    }
    const float4* vr = (const float4*)(vector_rep + (size_t)base * 48);
    #pragma unroll
    for (int it = 0; it < 6; ++it) {
      int idx = lane + 32 * it;           // 0..191
      int n = idx / 12, q = idx % 12;
      float4 v = vr[n * 12 + q];
      *(float4*)(V + n * VSTR + q * 4) = v;
    }
    for (int idx = lane; idx < 144; idx += 32) {
      int n = idx / 9, j = idx % 9;
      float c = cnt[base + n];
      c = fmaxf(c, 1.0f);
      M[n * MSTR + 144 + j] = sums[(size_t)(base + n) * 9 + j] / c;
    }
    for (int idx = lane; idx < 112; idx += 32) {
      int n = idx / 7, c = 153 + idx % 7;
      M[n * MSTR + c] = 0.0f;
    }
  }
  __syncthreads();

  // -------- vh = v_pre @ W_down^T ; vnorm -> merged[:,128:144] --------
  float vh[3][8];
  if (active) {
    #pragma unroll
    for (int t = 0; t < 3; ++t)
      #pragma unroll
      for (int hh = 0; hh < 8; ++hh) vh[t][hh] = 0.0f;
    const float* vn = V + lmod * VSTR;    // this lane's node, [V][3]
    #pragma unroll
    for (int vv = 0; vv < 16; ++vv) {
      float x = vn[vv * 3 + 0], y = vn[vv * 3 + 1], z = vn[vv * 3 + 2];
      #pragma unroll
      for (int hh = 0; hh < 8; ++hh) {
        float w = wd_lds[(lhalf * 8 + hh) * 16 + vv];
        vh[0][hh] += x * w; vh[1][hh] += y * w; vh[2][hh] += z * w;
      }
    }
    #pragma unroll
    for (int hh = 0; hh < 8; ++hh) {
      float s = vh[0][hh] * vh[0][hh] + vh[1][hh] * vh[1][hh] +
                vh[2][hh] * vh[2][hh] + 1e-8f;
      M[lmod * MSTR + 128 + lhalf * 8 + hh] = sqrtf(s);
    }
  }
  __syncthreads();
  if (active) {  // stash vh into V (reuse) as [n][t*16+h]
    #pragma unroll
    for (int t = 0; t < 3; ++t)
      #pragma unroll
      for (int hh = 0; hh < 8; ++hh)
        V[lmod * VSTR + t * 16 + lhalf * 8 + hh] = vh[t][hh];
  }

  // -------- Phase B: D[16 nodes x 128] = merged @ W_scalar^T via WMMA ------
  v8f acc[8];
  if (active) {
    #pragma unroll
    for (int t = 0; t < 8; ++t) {
      float b = b_scalar[t * 16 + lmod];
      #pragma unroll
      for (int r = 0; r < 8; ++r) acc[t][r] = b;
    }
    const float* Mrow = M + lmod * MSTR + 2 * lhalf;  // A[M=lmod][K=r+2*lhalf+k0]
    const float2* Bbase = (const float2*)Btp_scalar + lmod;
    for (int k0 = 0; k0 < KPAD; k0 += 4) {
      v2f a;
      a.x = Mrow[k0];
      a.y = Mrow[k0 + 1];
      // pair row index: (k0 + 2*lhalf)/2 ; each pair row = 128 float2
      const float2* Bp = Bbase + ((k0 >> 1) + lhalf) * 128;
      #pragma unroll
      for (int t = 0; t < 8; ++t) {
        float2 bb = Bp[t * 16];
        v2f b;
        b.x = bb.x;
        b.y = bb.y;
        acc[t] = __builtin_amdgcn_wmma_f32_16x16x4_f32(
            false, a, false, b, (short)0, acc[t], false, false);
      }
    }
    // SiLU; write s_final; stash silu into merged LDS for the gate GEMM
    #pragma unroll
    for (int t = 0; t < 8; ++t) {
      #pragma unroll
      for (int r = 0; r < 8; ++r) {
        float s = acc[t][r];
        float sil = s / (1.0f + __expf(-s));
        int node = r + 8 * lhalf;   // D layout: M = r + 8*(lane/16), N = lane%16
        s_final[(size_t)(base + node) * 128 + t * 16 + lmod] = sil;
        M[node * MSTR + t * 16 + lmod] = sil;
      }
    }
  }
  __syncthreads();

  // -------- gate = silu(s_out) @ W_gate^T + b_gate via WMMA ----------------
  if (active) {
    v8f g;
    float bg = b_gate[lmod];
    #pragma unroll
    for (int r = 0; r < 8; ++r) g[r] = bg;
    const float* Mrow = M + lmod * MSTR + 2 * lhalf;
    const float2* Bgbase = (const float2*)Btp_gate + lmod;
    for (int k0 = 0; k0 < 128; k0 += 4) {
      v2f a;
      a.x = Mrow[k0];
      a.y = Mrow[k0 + 1];
      float2 bb = Bgbase[((k0 >> 1) + lhalf) * 16];
      v2f b;
      b.x = bb.x;
      b.y = bb.y;
      g = __builtin_amdgcn_wmma_f32_16x16x4_f32(
          false, a, false, b, (short)0, g, false, false);
    }
    #pragma unroll
    for (int r = 0; r < 8; ++r) {
      int node = r + 8 * lhalf;
      G[node * GSTR + lmod] = 1.0f / (1.0f + __expf(-g[r]));
    }
  }
  __syncthreads();

  // -------- v_final = (vh @ W_up^T)^T * sigmoid(gate) ----------------------
  if (active) {
    float vout[24];
    const float* vhn = V + lmod * VSTR;   // [t*16+h] for node lmod
    #pragma unroll
    for (int vv = 0; vv < 8; ++vv) {
      int vo = lhalf * 8 + vv;
      float sg = G[lmod * GSTR + vo];
      #pragma unroll
      for (int t = 0; t < 3; ++t) {
        float s = 0.0f;
        #pragma unroll
        for (int h = 0; h < 16; ++h) s += vhn[t * 16 + h] * wu_lds[vo * 16 + h];
        vout[vv * 3 + t] = s * sg;
      }
    }
    float4* vp = (float4*)(v_final + (size_t)(base + lmod) * 48 + lhalf * 24);
    #pragma unroll
    for (int q = 0; q < 6; ++q)
      vp[q] = make_float4(vout[q * 4 + 0], vout[q * 4 + 1],
                          vout[q * 4 + 2], vout[q * 4 + 3]);
  }
}

// ---------------- host side ------------------------------------------------
extern "C" void kernel_launch(void* const* d_in, const int* in_sizes, int n_in,
                              void* d_out, int out_size, void* d_ws, size_t ws_size,
                              hipStream_t stream) {
  const float* scalar_rep = (const float*)d_in[0];
  const float* vector_rep = (const float*)d_in[1];
  const float* frames     = (const float*)d_in[2];
  const float* W_down     = (const float*)d_in[3];
  const float* W_frames   = (const float*)d_in[4];
  const float* W_scalar   = (const float*)d_in[5];
  const float* b_scalar   = (const float*)d_in[6];
  const float* W_up       = (const float*)d_in[7];
  const float* W_gate     = (const float*)d_in[8];
  const float* b_gate     = (const float*)d_in[9];
  const long long* edge_index = (const long long*)d_in[10];  // int64 [2][E]

  const int N = in_sizes[0] / 128;
  const int E = in_sizes[10] / 2;

  float* ws = (float*)d_ws;
  float* vdf        = ws;                         // N*9
  float* sums       = vdf + (size_t)N * 9;        // N*9
  float* cnt        = sums + (size_t)N * 9;       // N
  float* Btp_scalar = cnt + ((N + 3) & ~3);       // 160*128 (pair-interleaved)
  float* Btp_gate   = Btp_scalar + KPAD * 128;    // 128*16  (pair-interleaved)

  float* s_final = (float*)d_out;
  float* v_final = s_final + (size_t)N * 128;

  gcp_prep_kernel<<<(KPAD * 128 + 255) / 256, 256, 0, stream>>>(
      W_scalar, W_gate, Btp_scalar, Btp_gate);
  gcp_vdf_kernel<<<(N + 255) / 256, 256, 0, stream>>>(
      vector_rep, W_frames, vdf, sums, cnt, N);
  gcp_edge_kernel<<<(E + 255) / 256, 256, 0, stream>>>(
      frames, edge_index, vdf, sums, cnt, E);

  int tiles = N / 16;                       // N == 100000 -> 6250 tiles
  int blocks = (tiles + WAVES - 1) / WAVES;
  gcp_node_kernel<<<blocks, 32 * WAVES, 0, stream>>>(
      scalar_rep, vector_rep, W_down, b_scalar, W_up, b_gate,
      sums, cnt, Btp_scalar, Btp_gate, s_final, v_final, N);
}